// RNTN_25168508355265
// MI455X (gfx1250) — compile-verified
//
#include <hip/hip_runtime.h>
#include <hip/hip_bf16.h>
#include <math.h>

// ---------------------------------------------------------------------------
// RNTN (Recursive Neural Tensor Network), full binary tree, fp32.
//   N_LEAVES=4096, N_NODES=8191, HID=128, CS=256, VOCAB=200000, OUT=5
// Level-synchronous: leaf gather, 12 level kernels (2048..1 nodes), output.
// Bilinear term (69 GFLOP fp32) via V_WMMA_F32_16X16X4_F32:
//   tmp2 = CS @ V_k^T   (WMMA K-dim = e, contiguous in V rows -> b64 B loads)
//   bil[m] = sum_d tmp2[m,d] * cs[m,d]
// 32-node tiles: 2 m-tiles share each B fragment -> 4 WMMAs per
// {2 ds_load_b64 + 2 global_load_b64}. V (33.5 MB) resident in 192 MB L2.
// ---------------------------------------------------------------------------

#define HID      128
#define CSD      256          // 2*HID
#define OUT      5
#define N_LEAVES 4096
#define N_NODES  8191
#define TILE     32           // nodes per workgroup (2 WMMA m-tiles)
#define KGRP     8            // k (hidden-output) values per workgroup
#define LDS_STRIDE 260        // 256 + 4 pad floats: A-frag rows hit distinct banks

typedef __attribute__((ext_vector_type(2))) float v2f;
typedef __attribute__((ext_vector_type(8))) float v8f;

// ---------------------------------------------------------------------------
// Leaves: h[i] = embed[word_idx[i]]   (i = 0..4095, 128 floats each)
// ---------------------------------------------------------------------------
__global__ __launch_bounds__(128)
void rntn_leaf_kernel(float* __restrict__ h,
                      const float* __restrict__ embed,
                      const int* __restrict__ word_idx)
{
    const int i = blockIdx.x;
    const int t = threadIdx.x;
    const int w = word_idx[i];
    h[(size_t)i * HID + t] = embed[(size_t)w * HID + t];
}

// ---------------------------------------------------------------------------
// One tree level: nodes [base, base+count).
// grid = (ceil(count/32), 128/KGRP), block = 256 (8 wave32)
// Wave w owns d-columns [32w, 32w+32) as 2 N-tiles; m-tiles 0/1 shared.
// ---------------------------------------------------------------------------
__global__ __launch_bounds__(256)
void rntn_level_kernel(float* __restrict__ h,
                       const float* __restrict__ V,     // [128][256][256]
                       const float* __restrict__ W,     // [256][128]
                       const float* __restrict__ b,     // [128]
                       const int* __restrict__ left,
                       const int* __restrict__ right,
                       int base, int count)
{
    __shared__ float cs_lds[TILE * LDS_STRIDE];   // ~33.3 KB
    __shared__ float w_lds[CSD];                  // W[:,k] for current k
    __shared__ float bil_lds[TILE];

    const int tid   = threadIdx.x;
    const int wave  = tid >> 5;          // 0..7
    const int lane  = tid & 31;          // wave32
    const int mrow  = lane & 15;         // M row (A) / N col (B,D)
    const int khalf = lane >> 4;         // which K pair this lane holds
    const int tile0 = blockIdx.x * TILE;
    const int kbase = blockIdx.y * KGRP;

    // ---- stage CS tile [32 x 256] into LDS (zeros past count)
    {
        const int r  = tid >> 3;           // 0..31 row
        const int c0 = (tid & 7) * 32;     // 32-col chunk; never straddles 128
        const int node = tile0 + r;
        float4* dst = (float4*)&cs_lds[r * LDS_STRIDE + c0];
        if (node < count) {
            const int gi = base + node;
            const int ch = (c0 < HID) ? left[gi] : right[gi];
            const float4* src =
                (const float4*)(h + (size_t)ch * HID + (c0 & (HID - 1)));
            #pragma unroll
            for (int c = 0; c < 8; ++c) dst[c] = src[c];
        } else {
            #pragma unroll
            for (int c = 0; c < 8; ++c) dst[c] = make_float4(0.f, 0.f, 0.f, 0.f);
        }
    }
    __syncthreads();

    for (int kk = 0; kk < KGRP; ++kk) {
        const int k = kbase + kk;
        if (tid < TILE) bil_lds[tid] = 0.0f;
        w_lds[tid] = W[(size_t)tid * HID + k];       // blockDim == CSD
        __syncthreads();

        const float* __restrict__ Vk = V + (size_t)k * CSD * CSD;
        const int d0 = wave * 32;                    // this wave's d-columns

        // A frags: lane holds cs[mt*16+mrow, ks + 2*khalf + {0,1}]
        const float* ap0 = &cs_lds[(0 * 16 + mrow) * LDS_STRIDE + khalf * 2];
        const float* ap1 = &cs_lds[(1 * 16 + mrow) * LDS_STRIDE + khalf * 2];
        // B frags (K = e contiguous): lane holds V_k[d0+nt*16+mrow, ks+2*khalf+{0,1}]
        const float* bp0 = Vk + (size_t)(d0 + 0 * 16 + mrow) * CSD + khalf * 2;
        const float* bp1 = Vk + (size_t)(d0 + 1 * 16 + mrow) * CSD + khalf * 2;

        v8f acc00 = {}, acc01 = {}, acc10 = {}, acc11 = {};
        #pragma unroll 4
        for (int ks = 0; ks < CSD; ks += 4) {
            const v2f a0 = *(const v2f*)(ap0 + ks);
            const v2f a1 = *(const v2f*)(ap1 + ks);
            const v2f b0 = *(const v2f*)(bp0 + ks);
            const v2f b1 = *(const v2f*)(bp1 + ks);
            acc00 = __builtin_amdgcn_wmma_f32_16x16x4_f32(false, a0, false, b0,
                        (short)0, acc00, false, false);
            acc01 = __builtin_amdgcn_wmma_f32_16x16x4_f32(false, a0, false, b1,
                        (short)0, acc01, false, false);
            acc10 = __builtin_amdgcn_wmma_f32_16x16x4_f32(false, a1, false, b0,
                        (short)0, acc10, false, false);
            acc11 = __builtin_amdgcn_wmma_f32_16x16x4_f32(false, a1, false, b1,
                        (short)0, acc11, false, false);
        }

        // D layout: VGPR j -> (M = j + 8*khalf, N = mrow). Combine both
        // n-tiles per m before the wave32 shuffle tree (masks 1,2,4,8 stay
        // within each 16-lane half), then ds_add_f32 into bil_lds.
        #pragma unroll
        for (int mt = 0; mt < 2; ++mt) {
            const v8f* da = (mt == 0) ? &acc00 : &acc10;
            const v8f* db = (mt == 0) ? &acc01 : &acc11;
            float partial[8];
            #pragma unroll
            for (int j = 0; j < 8; ++j) {
                const int m = mt * 16 + j + 8 * khalf;
                partial[j] =
                    (*da)[j] * cs_lds[m * LDS_STRIDE + d0 + 0 * 16 + mrow] +
                    (*db)[j] * cs_lds[m * LDS_STRIDE + d0 + 1 * 16 + mrow];
            }
            #pragma unroll
            for (int j = 0; j < 8; ++j) {
                float p = partial[j];
                p += __shfl_xor(p, 1, 32);
                p += __shfl_xor(p, 2, 32);
                p += __shfl_xor(p, 4, 32);
                p += __shfl_xor(p, 8, 32);
                partial[j] = p;
            }
            if (mrow == 0) {
                #pragma unroll
                for (int j = 0; j < 8; ++j)
                    atomicAdd(&bil_lds[mt * 16 + j + 8 * khalf], partial[j]);
            }
        }

        // cs . W[:,k] : 256 threads, each a 32-wide chunk of one row.
        {
            const int r   = tid & 31;
            const int c0w = (tid >> 5) * 32;
            float wp = 0.0f;
            #pragma unroll 8
            for (int j = 0; j < 32; ++j)
                wp += cs_lds[r * LDS_STRIDE + c0w + j] * w_lds[c0w + j];
            atomicAdd(&bil_lds[r], wp);
        }
        __syncthreads();

        if (tid < TILE) {
            const int node = tile0 + tid;
            if (node < count)
                h[(size_t)(base + node) * HID + k] = tanhf(bil_lds[tid] + b[k]);
        }
        __syncthreads();
    }
}

// ---------------------------------------------------------------------------
// out = log_softmax(h @ W_out)  over all 8191 nodes. One wave32 per node.
// ---------------------------------------------------------------------------
__global__ __launch_bounds__(256)
void rntn_output_kernel(const float* __restrict__ h,
                        const float* __restrict__ W_out,  // [128][5]
                        float* __restrict__ out,
                        int n_nodes)
{
    const int node = blockIdx.x * 8 + (threadIdx.x >> 5);
    const int lane = threadIdx.x & 31;
    if (node >= n_nodes) return;

    const float* hr = h + (size_t)node * HID;
    float logits[OUT];
    #pragma unroll
    for (int o = 0; o < OUT; ++o) {
        float s = 0.0f;
        #pragma unroll
        for (int d = lane; d < HID; d += 32)
            s += hr[d] * W_out[(size_t)d * OUT + o];
        s += __shfl_xor(s, 1, 32);
        s += __shfl_xor(s, 2, 32);
        s += __shfl_xor(s, 4, 32);
        s += __shfl_xor(s, 8, 32);
        s += __shfl_xor(s, 16, 32);
        logits[o] = s;
    }
    if (lane == 0) {
        float mx = logits[0];
        #pragma unroll
        for (int o = 1; o < OUT; ++o) mx = fmaxf(mx, logits[o]);
        float se = 0.0f;
        #pragma unroll
        for (int o = 0; o < OUT; ++o) se += __expf(logits[o] - mx);
        const float lse = mx + __logf(se);
        #pragma unroll
        for (int o = 0; o < OUT; ++o)
            out[(size_t)node * OUT + o] = logits[o] - lse;
    }
}

// ---------------------------------------------------------------------------
// kernel_launch: graph-capture safe. d_ws holds h [8191 x 128] fp32 (~4.2 MB),
// fully rewritten every call -> deterministic.
// ---------------------------------------------------------------------------
extern "C" void kernel_launch(void* const* d_in, const int* in_sizes, int n_in,
                              void* d_out, int out_size, void* d_ws, size_t ws_size,
                              hipStream_t stream) {
    (void)in_sizes; (void)n_in; (void)out_size; (void)ws_size;

    const float* embed    = (const float*)d_in[0];   // [200000,128]
    const float* V        = (const float*)d_in[1];   // [128,256,256]
    const float* W        = (const float*)d_in[2];   // [256,128]
    const float* b        = (const float*)d_in[3];   // [1,128]
    const float* W_out    = (const float*)d_in[4];   // [128,5]
    // d_in[5] = is_leaf (implicit: nodes 0..4095 are leaves)
    const int*   word_idx = (const int*)d_in[6];     // [8191]
    const int*   left     = (const int*)d_in[7];     // [8191]
    const int*   right    = (const int*)d_in[8];     // [8191]
    float*       out      = (float*)d_out;           // [8191,5]

    float* h = (float*)d_ws;                         // [8191,128] scratch

    rntn_leaf_kernel<<<N_LEAVES, HID, 0, stream>>>(h, embed, word_idx);

    int base = N_LEAVES;
    for (int count = N_LEAVES / 2; count >= 1; count >>= 1) {
        dim3 grid((count + TILE - 1) / TILE, HID / KGRP);
        rntn_level_kernel<<<grid, 256, 0, stream>>>(h, V, W, b, left, right,
                                                    base, count);
        base += count;
    }

    rntn_output_kernel<<<(N_NODES + 7) / 8, 256, 0, stream>>>(h, W_out, out,
                                                              N_NODES);
}